// LieFlow_8589934818
// MI455X (gfx1250) — compile-verified
//
#include <hip/hip_runtime.h>
#include <hip/hip_bf16.h>

#define IN_FEATURES 1024
#define NBLOCKS     256
#define NLAYERS     17
#define BATCH       16384
#define ROWS_PER_WG 16
#define LSTRIDE     1028      // padded row stride (floats): banks = (4*row + f) % 64, distinct for row 0..15
#define WG_THREADS  128       // 4 wave32s; each wave owns 16 of the 64 feature tiles

typedef __attribute__((ext_vector_type(2))) float v2f;
typedef __attribute__((ext_vector_type(4))) float v4f;
typedef __attribute__((ext_vector_type(8))) float v8f;

// ---------------------------------------------------------------------------
// Kernel 1: expm of all 17*256 4x4 blocks via scaling-and-squaring Taylor.
// Also accumulates the batch-independent logdet constant -sum(trace(v)) and
// writes a 1024-entry identity permutation (used by layer 0's gather so the
// main kernel's inner loop is branchless).
// ---------------------------------------------------------------------------
__global__ __launch_bounds__(256) void expm4x4_kernel(const float* __restrict__ vs,
                                                      float* __restrict__ E,
                                                      float* __restrict__ traceAcc,
                                                      int*   __restrict__ ident) {
    int t = blockIdx.x * blockDim.x + threadIdx.x;
    if (t < IN_FEATURES) ident[t] = t;
    if (t >= NLAYERS * NBLOCKS) return;
    const float* v = vs + (size_t)t * 16;

    float A[16], T[16], P[16];
    float tr = v[0] + v[5] + v[10] + v[15];
#pragma unroll
    for (int i = 0; i < 16; ++i) A[i] = v[i] * 0.0625f;   // scale by 1/16
#pragma unroll
    for (int i = 0; i < 16; ++i) T[i] = (i == 0 || i == 5 || i == 10 || i == 15) ? 1.0f : 0.0f;

    // Horner Taylor: T = I + A*T/j, j = 8..1  => exp(A/16) to ~1e-14
    for (int j = 8; j >= 1; --j) {
        float inv = 1.0f / (float)j;
#pragma unroll
        for (int r = 0; r < 4; ++r)
#pragma unroll
            for (int c = 0; c < 4; ++c) {
                float s = 0.0f;
#pragma unroll
                for (int k = 0; k < 4; ++k) s += A[r * 4 + k] * T[k * 4 + c];
                P[r * 4 + c] = s;
            }
#pragma unroll
        for (int i = 0; i < 16; ++i)
            T[i] = P[i] * inv + ((i == 0 || i == 5 || i == 10 || i == 15) ? 1.0f : 0.0f);
    }
    // square 4 times: exp(A) = (exp(A/16))^16
    for (int sq = 0; sq < 4; ++sq) {
#pragma unroll
        for (int r = 0; r < 4; ++r)
#pragma unroll
            for (int c = 0; c < 4; ++c) {
                float s = 0.0f;
#pragma unroll
                for (int k = 0; k < 4; ++k) s += T[r * 4 + k] * T[k * 4 + c];
                P[r * 4 + c] = s;
            }
#pragma unroll
        for (int i = 0; i < 16; ++i) T[i] = P[i];
    }
#pragma unroll
    for (int i = 0; i < 16; ++i) E[(size_t)t * 16 + i] = T[i];

    atomicAdd(traceAcc, -tr);   // global_atomic_add_f32; ws zeroed by hipMemsetAsync
}

// ---------------------------------------------------------------------------
// Kernel 2: fully fused 17-layer flow. One WG = 16 rows resident in LDS
// (double buffered). Block-diagonal matmul runs on v_wmma_f32_16x16x4_f32:
// a 16-row x 16-feature tile = 4 chained K=4 WMMAs against a block-diagonal
// 16x16 B built from 4 consecutive expm blocks. The layer permutation is
// folded into the A-fragment gather addresses (bent-identity is elementwise
// and its logdet sum is permutation invariant). Layer 0 gathers through an
// identity table so the loop is branch-free.
// ---------------------------------------------------------------------------
__global__ __launch_bounds__(WG_THREADS) void lieflow_fused(const float* __restrict__ data,
                                                            const float* __restrict__ bsAll,
                                                            const int*   __restrict__ idxs,
                                                            const float* __restrict__ Ews,
                                                            const float* __restrict__ traceAcc,
                                                            const int*   __restrict__ ident,
                                                            float* __restrict__ out) {
    extern __shared__ float smem[];
    float* buf0 = smem;                             // 16 * 1028 floats
    float* buf1 = smem + ROWS_PER_WG * LSTRIDE;     // 16 * 1028 floats
    float* ldsh = smem + 2 * ROWS_PER_WG * LSTRIDE; // 16 floats (per-row logdet)

    const int tid  = threadIdx.x;
    const int lane = tid & 31;
    const int wave = tid >> 5;
    const int row0 = blockIdx.x * ROWS_PER_WG;
    const int h    = lane >> 4;     // WMMA lane half
    const int mN   = lane & 15;     // M for A loads, N for B/D

    if (tid < ROWS_PER_WG) ldsh[tid] = 0.0f;

    // ---- stage the 16-row tile of z into LDS (coalesced, non-temporal) ----
    for (int e4 = tid; e4 < ROWS_PER_WG * IN_FEATURES / 4; e4 += WG_THREADS) {
        int row = e4 >> 8;          // 256 float4 per row
        int f4  = e4 & 255;
        const v4f* src = (const v4f*)(data + (size_t)(row0 + row) * IN_FEATURES) + f4;
        v4f val = __builtin_nontemporal_load(src);
        *(v4f*)(buf0 + row * LSTRIDE + f4 * 4) = val;  // 1028*4 B row stride: 16B aligned
    }

    float* cur = buf0;
    float* nxt = buf1;
    const int* gather = ident;      // layer 0 reads unpermuted data
    float ldacc = 0.0f;

    for (int l = 0; l < NLAYERS; ++l) {
        __syncthreads();            // cur fully written (init load / previous bent)
        const float* El = Ews   + (size_t)l * NBLOCKS * 16;
        const float* bl = bsAll + (size_t)l * IN_FEATURES;

        // ---- lie layer: each wave computes 16 tiles of 16 rows x 16 features ----
        for (int i = 0; i < 16; ++i) {
            const int fb = (wave * 16 + i) * 16;     // feature base of this tile
            v8f acc = {0.0f, 0.0f, 0.0f, 0.0f, 0.0f, 0.0f, 0.0f, 0.0f};
#pragma unroll
            for (int k = 0; k < 4; ++k) {
                // A slice k: inputs = gathered features fb+4k .. fb+4k+3
                int e0 = fb + 4 * k + 2 * h;
                int s0 = gather[e0];
                int s1 = gather[e0 + 1];
                v2f a;
                a.x = cur[mN * LSTRIDE + s0];
                a.y = cur[mN * LSTRIDE + s1];
                // B slice k: rows 4k..4k+3 of block-diagonal 16x16 E-matrix
                const float* Eb = El + ((fb >> 2) + k) * 16;
                float ex = Eb[(2 * h + 0) * 4 + (mN & 3)];
                float ey = Eb[(2 * h + 1) * 4 + (mN & 3)];
                bool sel = ((mN >> 2) == k);
                v2f b;
                b.x = sel ? ex : 0.0f;
                b.y = sel ? ey : 0.0f;
                acc = __builtin_amdgcn_wmma_f32_16x16x4_f32(
                    false, a, false, b, (short)0, acc, false, false);
            }
            float bv = bl[fb + mN];
#pragma unroll
            for (int r = 0; r < 8; ++r)
                nxt[(r + 8 * h) * LSTRIDE + fb + mN] = acc[r] + bv;
        }
        __syncthreads();            // nxt fully written

        // ---- bent identity (skipped after the final lie layer) ----
        if (l < NLAYERS - 1) {
            int row = tid >> 3;                 // 8 threads per row
            int f0  = (tid & 7) * 128;
            float* zp = nxt + row * LSTRIDE + f0;
            float sacc = 0.0f;
            for (int j = 0; j < 128; ++j) {
                float z = zp[j];
                float s = __builtin_sqrtf(z * z + 1.0f);
                sacc += __logf(z / (2.0f * s) + 1.0f);
                zp[j] = (s - 1.0f) * 0.5f + z;
            }
            ldacc -= sacc;
        }
        gather = idxs + (size_t)l * IN_FEATURES;
        float* t = cur; cur = nxt; nxt = t;
    }

    // ---- final stores: apply last permutation while writing z out (NT) ----
    for (int e = tid; e < ROWS_PER_WG * IN_FEATURES; e += WG_THREADS) {
        int row = e >> 10, f = e & 1023;
        float zval = cur[row * LSTRIDE + gather[f]];
        __builtin_nontemporal_store(zval, out + (size_t)(row0 + row) * IN_FEATURES + f);
    }
    atomicAdd(&ldsh[tid >> 3], ldacc);          // ds_add_f32 per-row reduction
    __syncthreads();
    if (tid < ROWS_PER_WG) {
        const size_t Z = (size_t)BATCH * IN_FEATURES;
        out[Z + row0 + tid] = ldsh[tid] + traceAcc[0];
    }
}

// ---------------------------------------------------------------------------
extern "C" void kernel_launch(void* const* d_in, const int* in_sizes, int n_in,
                              void* d_out, int out_size, void* d_ws, size_t ws_size,
                              hipStream_t stream) {
    const float* data = (const float*)d_in[0];
    const float* vs   = (const float*)d_in[1];
    const float* bs   = (const float*)d_in[2];
    const int*   idxs = (const int*)d_in[3];
    float* out = (float*)d_out;

    float* traceAcc = (float*)d_ws;
    float* Ews      = (float*)((char*)d_ws + 256);
    int*   ident    = (int*)((char*)d_ws + 256 + (size_t)NLAYERS * NBLOCKS * 16 * sizeof(float));

    // zero the trace accumulator (memset nodes are graph-capturable)
    (void)hipMemsetAsync(d_ws, 0, 256, stream);

    expm4x4_kernel<<<(NLAYERS * NBLOCKS + 255) / 256, 256, 0, stream>>>(vs, Ews, traceAcc, ident);

    size_t shmem = (size_t)(2 * ROWS_PER_WG * LSTRIDE + 32) * sizeof(float); // ~129 KB of 320 KB LDS
    (void)hipFuncSetAttribute((const void*)lieflow_fused,
                              hipFuncAttributeMaxDynamicSharedMemorySize, (int)shmem);
    lieflow_fused<<<BATCH / ROWS_PER_WG, WG_THREADS, shmem, stream>>>(
        data, bs, idxs, Ews, traceAcc, ident, out);
}